// FilterbankLinear_26645977104526
// MI455X (gfx1250) — compile-verified
//
#include <hip/hip_runtime.h>

typedef float v2f __attribute__((ext_vector_type(2)));
typedef float v8f __attribute__((ext_vector_type(8)));

#define BATCH     32
#define IN_CH     21
#define IN_FEAT   4096
#define WIN       32
#define OUT_CH    8
#define N_WIN     4064                 // int((4096-32)/1 - 1e-7) + 1
#define NK        (OUT_CH * N_WIN)     // 32512
#define DDIM      (IN_CH * WIN)        // 672

// One wave (32 lanes) computes the full 32x8 output tile for one window j:
//   C(32x8) = A(32x672) x W(672x8)
// via 168 x V_WMMA_F32_16X16X4_F32 steps with two accumulators (batch halves).
__global__ __launch_bounds__(256)
void FilterbankLinear_kernel(const float* __restrict__ x,
                             const float* __restrict__ w,
                             float* __restrict__ out) {
    const int lane = threadIdx.x & 31;
    const int wave = threadIdx.x >> 5;
    const int j = blockIdx.x * 8 + wave;
    if (j >= N_WIN) return;                       // whole-wave guard (EXEC stays full)

    const int start = (j == N_WIN - 1) ? (IN_FEAT - WIN) : j;

    const int r    = lane & 15;   // A: M row (batch % 16); B/C/D: N column
    const int half = lane >> 4;   // selects K pair within the 4-chunk
    const int koff = half * 2;    // k offset {0,2}

    // A-operand base pointers: batch rows r (acc0) and r+16 (acc1)
    const float* __restrict__ xlo = x + (size_t)r        * (IN_CH * IN_FEAT);
    const float* __restrict__ xhi = x + (size_t)(r + 16) * (IN_CH * IN_FEAT);

    // B-operand: weight row for out-channel n = r; lanes n>=8 clamp in-bounds
    // and are zero-masked in registers (cndmask, no branch in the loop).
    const int  n      = r;
    const bool nvalid = (n < OUT_CH);
    const int  nclamp = nvalid ? n : (OUT_CH - 1);
    const float* __restrict__ wrow =
        w + (size_t)(j * OUT_CH + nclamp) * DDIM + koff;  // even offset -> 8B aligned

    v8f acc0 = {};   // batches 0..15
    v8f acc1 = {};   // batches 16..31

    #pragma unroll 8
    for (int kb = 0; kb < DDIM; kb += 4) {
        // 4|32 => a k-chunk never crosses an input-channel boundary
        const int c = kb >> 5;                  // input channel (scalar)
        const int t = (kb & 31) + koff;         // position within window (per-half)
        const int p = start + t;                // feature position

        const float* alo = xlo + c * IN_FEAT + p;
        const float* ahi = xhi + c * IN_FEAT + p;
        v2f a0; a0.x = alo[0]; a0.y = alo[1];   // K-pair for batch half 0
        v2f a1; a1.x = ahi[0]; a1.y = ahi[1];   // K-pair for batch half 1

        // Weights are read exactly once overall -> non-temporal (don't pollute L2)
        v2f bv = __builtin_nontemporal_load((const v2f*)(wrow + kb));
        bv.x = nvalid ? bv.x : 0.0f;            // zero-pad N columns 8..15
        bv.y = nvalid ? bv.y : 0.0f;

        acc0 = __builtin_amdgcn_wmma_f32_16x16x4_f32(
                   false, a0, false, bv, (short)0, acc0, false, false);
        acc1 = __builtin_amdgcn_wmma_f32_16x16x4_f32(
                   false, a1, false, bv, (short)0, acc1, false, false);
    }

    // D layout: lane = n + 16*(m>=8), VGPR v = m&7. Store only valid channels.
    if (nvalid) {
        float* obase = out + (size_t)j * OUT_CH + n;
        #pragma unroll
        for (int v = 0; v < 8; ++v) {
            const int m = v + half * 8;
            obase[(size_t)m        * NK] = acc0[v];
            obase[(size_t)(m + 16) * NK] = acc1[v];
        }
    }
}

extern "C" void kernel_launch(void* const* d_in, const int* in_sizes, int n_in,
                              void* d_out, int out_size, void* d_ws, size_t ws_size,
                              hipStream_t stream) {
    (void)in_sizes; (void)n_in; (void)out_size; (void)d_ws; (void)ws_size;
    const float* x = (const float*)d_in[0];   // (32, 21, 4096) fp32
    const float* w = (const float*)d_in[1];   // (32512, 672) fp32
    float* out = (float*)d_out;               // (32, 32512) fp32

    dim3 grid(N_WIN / 8);   // 508 blocks x 8 waves = 4064 waves, one per window
    dim3 block(256);
    FilterbankLinear_kernel<<<grid, block, 0, stream>>>(x, w, out);
}